// ExtraMSAEmbedding_44375602102636
// MI455X (gfx1250) — compile-verified
//
#include <hip/hip_runtime.h>

typedef __attribute__((ext_vector_type(2))) float v2f;
typedef __attribute__((ext_vector_type(8))) float v8f;

#define N_SEQ   2048
#define N_RES   384
#define C_IN    25
#define C_OUT   64
#define M_TOTAL (N_SEQ * N_RES)      // 786432 rows
#define M_TILES (M_TOTAL / 16)       // 49152
#define N_TILES (C_OUT / 16)         // 4
#define WAVES_PER_BLOCK 8
#define BLOCKS ((M_TILES * N_TILES) / WAVES_PER_BLOCK)  // 24576, exact

// One wave32 computes one 16x16 f32 output tile via 7x V_WMMA_F32_16X16X4_F32
// (K = 25 padded to 28). A-fragments are synthesized in registers from the
// one-hot class id + deletion scalars; nothing but the 201MB output is streamed.
__global__ __launch_bounds__(256) void extra_msa_embed_wmma(
    const int*   __restrict__ msa,      // [M_TOTAL] class ids 0..22
    const float* __restrict__ has_del,  // [M_TOTAL]
    const float* __restrict__ del_val,  // [M_TOTAL]
    const float* __restrict__ W,        // [C_OUT, C_IN]
    const float* __restrict__ bias,     // [C_OUT]
    float*       __restrict__ out)      // [M_TOTAL, C_OUT]
{
    const int wave  = threadIdx.x >> 5;
    const int lane  = threadIdx.x & 31;
    const int tile  = blockIdx.x * WAVES_PER_BLOCK + wave;
    const int mtile = tile >> 2;          // which 16-row strip
    const int ntile = tile & 3;           // which 16-col strip of the 64 outputs
    const int mbase = mtile << 4;
    const int nbase = ntile << 4;
    const int l15   = lane & 15;
    const int khalf = (lane >> 4) << 1;   // 0 for lanes 0-15, 2 for lanes 16-31

    // Per-lane row features (A side): row M = l15 of this tile.
    const int   m   = mbase + l15;
    const int   cls = msa[m];
    const float hd  = has_del[m];
    const float dv  = del_val[m];

    // Per-lane weight row (B side): column N = nbase + l15.
    const int    n    = nbase + l15;
    const float* Wrow = W + n * C_IN;

    v8f acc = {};
#pragma unroll
    for (int kb = 0; kb < 28; kb += 4) {   // K padded 25 -> 28
        v2f a, bf;
#pragma unroll
        for (int v = 0; v < 2; ++v) {
            const int k = kb + khalf + v;
            // A[m, k]: one-hot(cls) for k<23, then has_del, del_val, zero-pad.
            float av = (k < 23) ? ((cls == k) ? 1.0f : 0.0f)
                     : (k == 23) ? hd
                     : (k == 24) ? dv
                     : 0.0f;
            a[v] = av;
            // B[k, n] = W[n, k]; clamp index then value-select to avoid
            // divergent control flow around the load (keeps EXEC all-ones).
            const int   ks = (k < C_IN) ? k : 0;
            const float wv = Wrow[ks];
            bf[v] = (k < C_IN) ? wv : 0.0f;
        }
        // (neg_a, A, neg_b, B, c_mod, C, reuse_a, reuse_b)
        acc = __builtin_amdgcn_wmma_f32_16x16x4_f32(
            false, a, false, bf, (short)0, acc, false, false);
    }

    // C/D layout: lanes 0-15 -> rows v (N = l15), lanes 16-31 -> rows v+8.
    const float bn    = bias[n];
    const int   mrow0 = (lane < 16) ? 0 : 8;
#pragma unroll
    for (int v = 0; v < 8; ++v) {
        out[(size_t)(mbase + mrow0 + v) * C_OUT + n] = acc[v] + bn;
    }
}

extern "C" void kernel_launch(void* const* d_in, const int* in_sizes, int n_in,
                              void* d_out, int out_size, void* d_ws, size_t ws_size,
                              hipStream_t stream) {
    (void)in_sizes; (void)n_in; (void)out_size; (void)d_ws; (void)ws_size;
    const int*   msa = (const int*)  d_in[0];   // extra_msa        [2048,384] int32
    const float* hd  = (const float*)d_in[1];   // extra_has_deletion
    const float* dv  = (const float*)d_in[2];   // extra_deletion_value
    const float* W   = (const float*)d_in[3];   // [64,25]
    const float* b   = (const float*)d_in[4];   // [64]
    float*       out = (float*)d_out;           // [2048,384,64]

    extra_msa_embed_wmma<<<BLOCKS, 256, 0, stream>>>(msa, hd, dv, W, b, out);
}